// Encoder_13383118094555
// MI455X (gfx1250) — compile-verified
//
#include <hip/hip_runtime.h>

typedef __attribute__((ext_vector_type(2))) float v2f;
typedef __attribute__((ext_vector_type(8))) float v8f;

#define FDIM 512

// ---------------------------------------------------------------------------
// CSR construction: in-degree histogram, deg^{-1/2}, scan, cursor fill
// ---------------------------------------------------------------------------
__global__ void gcn_zero_cnt(int* __restrict__ cnt, int n) {
    int i = blockIdx.x * blockDim.x + threadIdx.x;
    if (i < n) cnt[i] = 0;
}

__global__ void gcn_count(const int* __restrict__ col, int* __restrict__ cnt, int E) {
    int e = blockIdx.x * blockDim.x + threadIdx.x;
    if (e < E) atomicAdd(&cnt[col[e]], 1);
}

__global__ void gcn_dis(const int* __restrict__ cnt, float* __restrict__ dis, int n) {
    int i = blockIdx.x * blockDim.x + threadIdx.x;
    if (i < n) dis[i] = rsqrtf((float)(cnt[i] + 1));  // +1 = self loop
}

// Single-workgroup exclusive scan over cnt -> rowPtr[0..n]
__global__ __launch_bounds__(1024) void gcn_scan(const int* __restrict__ cnt,
                                                 int* __restrict__ rowPtr, int n)
{
    __shared__ int buf[1024];
    __shared__ int carry;
    const int tid = threadIdx.x;
    if (tid == 0) { carry = 0; rowPtr[0] = 0; }
    __syncthreads();
    for (int base = 0; base < n; base += 1024) {
        int v = (base + tid < n) ? cnt[base + tid] : 0;
        buf[tid] = v;
        __syncthreads();
        for (int off = 1; off < 1024; off <<= 1) {   // Hillis-Steele inclusive
            int t = (tid >= off) ? buf[tid - off] : 0;
            __syncthreads();
            buf[tid] += t;
            __syncthreads();
        }
        if (base + tid < n) rowPtr[base + tid + 1] = carry + buf[tid];
        __syncthreads();
        if (tid == 0) carry += buf[1023];
        __syncthreads();
    }
}

__global__ void gcn_cursor_init(const int* __restrict__ rowPtr, int* __restrict__ cur, int n) {
    int i = blockIdx.x * blockDim.x + threadIdx.x;
    if (i < n) cur[i] = rowPtr[i];
}

__global__ void gcn_fill(const int* __restrict__ row, const int* __restrict__ col,
                         int* __restrict__ cur, int* __restrict__ csrSrc, int E)
{
    int e = blockIdx.x * blockDim.x + threadIdx.x;
    if (e < E) {
        int slot = atomicAdd(&cur[col[e]], 1);
        csrSrc[slot] = row[e];
    }
}

// ---------------------------------------------------------------------------
// WMMA fp32 GEMM: H[M,512] = X[M,512] @ W[512,512]
// One 256-thread block (8 wave32) computes a 16-row x 512-col stripe.
// Wave w owns cols [64w, 64w+64): four 16x16 f32 accumulators.
// Safe in-place (X == H): barrier before stores.
// ---------------------------------------------------------------------------
__global__ __launch_bounds__(256) void gcn_gemm_f32(
    const float* __restrict__ X, const float* __restrict__ W,
    float* __restrict__ H, int nRows)
{
    const int wave = threadIdx.x >> 5;
    const int lane = threadIdx.x & 31;
    const int l15  = lane & 15;
    const int hi   = lane >> 4;
    const int row0 = blockIdx.x << 4;
    const int colBase = wave << 6;

    int m  = row0 + l15;
    int mc = (m < nRows) ? m : (nRows - 1);
    const float* xrow = X + (size_t)mc * FDIM;

    v8f acc0 = {}, acc1 = {}, acc2 = {}, acc3 = {};
    const int n0 = colBase + l15;

    for (int k = 0; k < FDIM; k += 4) {
        const int ka = k + (hi << 1);
        v2f a;
        a.x = xrow[ka];
        a.y = xrow[ka + 1];

        const float* wr0 = W + (size_t)ka * FDIM;
        const float* wr1 = wr0 + FDIM;
        v2f b0, b1, b2, b3;
        b0.x = wr0[n0];      b0.y = wr1[n0];
        b1.x = wr0[n0 + 16]; b1.y = wr1[n0 + 16];
        b2.x = wr0[n0 + 32]; b2.y = wr1[n0 + 32];
        b3.x = wr0[n0 + 48]; b3.y = wr1[n0 + 48];

        acc0 = __builtin_amdgcn_wmma_f32_16x16x4_f32(false, a, false, b0, (short)0, acc0, false, false);
        acc1 = __builtin_amdgcn_wmma_f32_16x16x4_f32(false, a, false, b1, (short)0, acc1, false, false);
        acc2 = __builtin_amdgcn_wmma_f32_16x16x4_f32(false, a, false, b2, (short)0, acc2, false, false);
        acc3 = __builtin_amdgcn_wmma_f32_16x16x4_f32(false, a, false, b3, (short)0, acc3, false, false);
    }

    __syncthreads();  // all A-reads complete before any in-place stores

    #pragma unroll
    for (int v = 0; v < 8; ++v) {
        int r = row0 + v + (hi << 3);
        if (r < nRows) {
            float* hr = H + (size_t)r * FDIM + colBase + l15;
            hr[0]  = acc0[v];
            hr[16] = acc1[v];
            hr[32] = acc2[v];
            hr[48] = acc3[v];
        }
    }
}

// ---------------------------------------------------------------------------
// Gather aggregation (no atomics):
// out[i,:] = b + H[i,:]*dis[i]^2 + sum_{e: col==i} H[src,:]*dis[src]*dis[i]
// One 128-thread block per node; each lane owns 4 contiguous features.
// ---------------------------------------------------------------------------
__global__ __launch_bounds__(128) void gcn_gather(
    const float* __restrict__ H, const int* __restrict__ rowPtr,
    const int* __restrict__ csrSrc, const float* __restrict__ dis,
    const float* __restrict__ b, float* __restrict__ out, int n)
{
    const int i   = blockIdx.x;
    const int tid = threadIdx.x;
    if (i >= n) return;

    const int start = rowPtr[i];
    const int end   = rowPtr[i + 1];
    const float di  = dis[i];

    // self-loop term
    float4 hv = ((const float4*)(H + (size_t)i * FDIM))[tid];
    const float ws = di * di;
    float4 acc;
    acc.x = hv.x * ws;
    acc.y = hv.y * ws;
    acc.z = hv.z * ws;
    acc.w = hv.w * ws;

    for (int e = start; e < end; ++e) {
        int s = csrSrc[e];
        int sNext = (e + 1 < end) ? csrSrc[e + 1] : s;
        __builtin_prefetch(H + (size_t)sNext * FDIM + (tid << 2), 0, 0);
        float w = dis[s] * di;
        float4 v = ((const float4*)(H + (size_t)s * FDIM))[tid];
        acc.x += v.x * w;
        acc.y += v.y * w;
        acc.z += v.z * w;
        acc.w += v.w * w;
    }

    float4 bv = ((const float4*)b)[tid];
    float4 o;
    o.x = acc.x + bv.x;
    o.y = acc.y + bv.y;
    o.z = acc.z + bv.z;
    o.w = acc.w + bv.w;
    ((float4*)(out + (size_t)i * FDIM))[tid] = o;
}

// ---------------------------------------------------------------------------
// H = relu(out)
// ---------------------------------------------------------------------------
__global__ void gcn_relu(const float* __restrict__ in, float* __restrict__ o, size_t total) {
    size_t idx = (size_t)blockIdx.x * blockDim.x + threadIdx.x;
    if (idx < total) o[idx] = fmaxf(in[idx], 0.0f);
}

// ---------------------------------------------------------------------------
extern "C" void kernel_launch(void* const* d_in, const int* in_sizes, int n_in,
                              void* d_out, int out_size, void* d_ws, size_t ws_size,
                              hipStream_t stream)
{
    const float* x   = (const float*)d_in[0];
    const int*   ei  = (const int*)d_in[1];   // [2,E] flat: row at 0, col at E
    const float* W1  = (const float*)d_in[2];
    const float* b1  = (const float*)d_in[3];
    const float* W2  = (const float*)d_in[4];
    const float* b2  = (const float*)d_in[5];
    float* out = (float*)d_out;

    const int N = in_sizes[0] / FDIM;
    const int E = in_sizes[1] / 2;
    const int* rowIdx = ei;
    const int* colIdx = ei + E;

    // workspace carve-out (256B aligned blocks)
    char* p = (char*)d_ws;
    auto alloc = [&](size_t bytes) {
        char* r = p;
        p += (bytes + 255) & ~(size_t)255;
        return r;
    };
    int*   cnt    = (int*)alloc((size_t)N * 4);
    int*   rowPtr = (int*)alloc((size_t)(N + 1) * 4);
    int*   cursor = (int*)alloc((size_t)N * 4);
    float* dis    = (float*)alloc((size_t)N * 4);
    int*   csrSrc = (int*)alloc((size_t)E * 4);
    const size_t total = (size_t)N * FDIM;
    float* H      = (float*)alloc(total * 4);

    const int eltBlocks  = (int)((total + 255) / 256);
    const int nBlocks    = (N + 255) / 256;
    const int eBlocks    = (E + 255) / 256;
    const int gemmBlocks = (N + 15) / 16;    // 3125 for N=50000

    // Build destination-CSR + normalization (once, reused by both layers)
    gcn_zero_cnt<<<nBlocks, 256, 0, stream>>>(cnt, N);
    gcn_count<<<eBlocks, 256, 0, stream>>>(colIdx, cnt, E);
    gcn_dis<<<nBlocks, 256, 0, stream>>>(cnt, dis, N);
    gcn_scan<<<1, 1024, 0, stream>>>(cnt, rowPtr, N);
    gcn_cursor_init<<<nBlocks, 256, 0, stream>>>(rowPtr, cursor, N);
    gcn_fill<<<eBlocks, 256, 0, stream>>>(rowIdx, colIdx, cursor, csrSrc, E);

    // Layer 1
    gcn_gemm_f32<<<gemmBlocks, 256, 0, stream>>>(x, W1, H, N);
    gcn_gather<<<N, 128, 0, stream>>>(H, rowPtr, csrSrc, dis, b1, out, N);

    // ReLU into H, then layer 2 (GEMM in place on H)
    gcn_relu<<<eltBlocks, 256, 0, stream>>>(out, H, total);
    gcn_gemm_f32<<<gemmBlocks, 256, 0, stream>>>(H, W2, H, N);
    gcn_gather<<<N, 128, 0, stream>>>(H, rowPtr, csrSrc, dis, b2, out, N);
}